// moe_layer_16741782520583
// MI455X (gfx1250) — compile-verified
//
#include <hip/hip_runtime.h>
#include <hip/hip_bf16.h>

#define HIDDEN 20
#define NEXP 5
#define WSTEPS 25                 // K = 100 (5 experts * 20) in slices of 4
#define NSTEPS 27                 // + 2 bias slices (K 100..107, rows >=105 zero)
#define GSTEPS 5                  // gating: K = 20 in slices of 4

typedef __attribute__((ext_vector_type(2))) float v2f;
typedef __attribute__((ext_vector_type(4))) float v4f;
typedef __attribute__((ext_vector_type(8))) float v8f;

__device__ __forceinline__ v8f wmma_f32(v2f a, v2f b, v8f c) {
    return __builtin_amdgcn_wmma_f32_16x16x4_f32(
        /*neg_a=*/false, a, /*neg_b=*/false, b,
        /*c_mod=*/(short)0, c, /*reuse_a=*/false, /*reuse_b=*/false);
}

__global__ __launch_bounds__(256) void moe_top1_wmma_kernel(
    const float* __restrict__ x,   // [T, 20]
    const float* __restrict__ wg,  // [20, 5]
    const float* __restrict__ We,  // [5, 20, 20]
    const float* __restrict__ be,  // [5, 20]
    float* __restrict__ out,       // [T, 20]
    int numTokens)
{
    // B fragments in WMMA lane layout: entry (s, j, lane) -> 2 floats (k, k+1)
    __shared__ float fragB[NSTEPS * 2 * 32 * 2];   // 13.5 KB
    __shared__ float fragG[GSTEPS * 32 * 2];       // 1.25 KB
    __shared__ float logitsT[8][16 * 16];          // per-wave transpose scratch, 8 KB

    const int tid  = threadIdx.x;
    const int lane = tid & 31;
    const int wave = tid >> 5;

    // ---- pack big-B (We stacked along K, bias rows appended) into fragments ----
    for (int idx = tid; idx < NSTEPS * 2 * 32; idx += blockDim.x) {
        const int s = idx >> 6;
        const int j = (idx >> 5) & 1;
        const int l = idx & 31;
        const int kbase = 4 * s + ((l < 16) ? 0 : 2);
        const int n = j * 16 + (l & 15);
        float v[2];
        #pragma unroll
        for (int t = 0; t < 2; ++t) {
            const int k = kbase + t;
            float val = 0.0f;
            if (n < HIDDEN) {
                if (k < NEXP * HIDDEN) {
                    const int e = k / HIDDEN, kl = k % HIDDEN;
                    val = We[e * HIDDEN * HIDDEN + kl * HIDDEN + n];
                } else if (k < NEXP * HIDDEN + NEXP) {
                    val = be[(k - NEXP * HIDDEN) * HIDDEN + n];
                }
            }
            v[t] = val;
        }
        fragB[idx * 2 + 0] = v[0];
        fragB[idx * 2 + 1] = v[1];
    }
    // ---- gate weight fragments (5 valid cols, rest zero) ----
    for (int idx = tid; idx < GSTEPS * 32; idx += blockDim.x) {
        const int s = idx >> 5;
        const int l = idx & 31;
        const int kbase = 4 * s + ((l < 16) ? 0 : 2);
        const int n = l & 15;
        float v0 = 0.0f, v1 = 0.0f;
        if (n < NEXP) {
            v0 = wg[(kbase + 0) * NEXP + n];
            v1 = wg[(kbase + 1) * NEXP + n];
        }
        fragG[idx * 2 + 0] = v0;
        fragG[idx * 2 + 1] = v1;
    }
    __syncthreads();

    const int numTiles = numTokens >> 4;
    const int waveId   = blockIdx.x * (blockDim.x >> 5) + wave;
    const int numWaves = gridDim.x * (blockDim.x >> 5);
    const int  m       = lane & 15;
    const bool hiHalf  = lane >= 16;

    for (int tile = waveId; tile < numTiles; tile += numWaves) {
        const int tokBase = tile << 4;

        // Each lane holds all 20 inputs of token (lane&15); halves duplicate.
        const float* xt = x + (size_t)(tokBase + m) * HIDDEN;
        v4f xv[5];
        #pragma unroll
        for (int q = 0; q < 5; ++q)
            xv[q] = *(const v4f*)(xt + 4 * q);

        // Prefetch next grid-stride tile (global_prefetch_b8).
        if (tile + numWaves < numTiles)
            __builtin_prefetch(x + (size_t)(tokBase + (numWaves << 4) + m) * HIDDEN, 0, 1);

        // ---- gating logits via WMMA: [16 tok x 20] @ [20 x 5(pad16)] ----
        v8f clog = {};
        #pragma unroll
        for (int s = 0; s < GSTEPS; ++s) {
            v2f a;
            a.x = xv[s][hiHalf ? 2 : 0];
            a.y = xv[s][hiHalf ? 3 : 1];
            v2f b = *(const v2f*)&fragG[(s * 32 + lane) * 2];
            clog = wmma_f32(a, b, clog);
        }
        // Transpose logits through per-wave LDS (DS ops are wave-ordered).
        float* sw = logitsT[wave];
        #pragma unroll
        for (int r = 0; r < 8; ++r)
            sw[(r + (hiHalf ? 8 : 0)) * 16 + m] = clog[r];
        float lg[NEXP];
        #pragma unroll
        for (int e = 0; e < NEXP; ++e)
            lg[e] = sw[m * 16 + e];

        // ---- top-1 softmax gate (argmax(prob) == argmax(logit)) ----
        float lmax = lg[0];
        int top1 = 0;
        #pragma unroll
        for (int e = 1; e < NEXP; ++e)
            if (lg[e] > lmax) { lmax = lg[e]; top1 = e; }
        float psum = 0.0f;
        #pragma unroll
        for (int e = 0; e < NEXP; ++e)
            psum += __expf(lg[e] - lmax);
        const float gate = 1.0f / psum;   // == softmax prob of the max logit

        // ---- routed expert matmul: masked-A over K-stacked experts ----
        v8f c0 = {}, c1 = {};
        #pragma unroll
        for (int s = 0; s < WSTEPS; ++s) {
            const int e  = s / 5;
            const int kl = s % 5;            // vector index into xv
            v2f a;
            const bool sel = (top1 == e);
            a.x = sel ? xv[kl][hiHalf ? 2 : 0] : 0.0f;
            a.y = sel ? xv[kl][hiHalf ? 3 : 1] : 0.0f;
            v2f b0 = *(const v2f*)&fragB[(((s * 2) + 0) * 32 + lane) * 2];
            v2f b1 = *(const v2f*)&fragB[(((s * 2) + 1) * 32 + lane) * 2];
            c0 = wmma_f32(a, b0, c0);
            c1 = wmma_f32(a, b1, c1);
        }
        // Bias slices: one-hot A rows select be[top1]. s=25: K100..103, s=26: K104..107.
        {
            v2f a;
            a.x = (top1 == (hiHalf ? 2 : 0)) ? 1.0f : 0.0f;
            a.y = (top1 == (hiHalf ? 3 : 1)) ? 1.0f : 0.0f;
            v2f b0 = *(const v2f*)&fragB[(((25 * 2) + 0) * 32 + lane) * 2];
            v2f b1 = *(const v2f*)&fragB[(((25 * 2) + 1) * 32 + lane) * 2];
            c0 = wmma_f32(a, b0, c0);
            c1 = wmma_f32(a, b1, c1);

            a.x = (!hiHalf && top1 == 4) ? 1.0f : 0.0f;   // K=104 -> expert 4
            a.y = 0.0f;                                    // K>=105 zero pad
            b0 = *(const v2f*)&fragB[(((26 * 2) + 0) * 32 + lane) * 2];
            b1 = *(const v2f*)&fragB[(((26 * 2) + 1) * 32 + lane) * 2];
            c0 = wmma_f32(a, b0, c0);
            c1 = wmma_f32(a, b1, c1);
        }

        // ---- epilogue: gate-scale and store (C layout: VGPR r -> rows r / r+8) ----
        #pragma unroll
        for (int r = 0; r < 8; ++r) {
            const int rowOff  = r + (hiHalf ? 8 : 0);
            const int tok     = tokBase + rowOff;
            const float g     = __shfl(gate, rowOff, 32);  // lane rowOff holds that token's gate
            float* op = out + (size_t)tok * HIDDEN;
            op[m] = g * c0[r];                 // cols 0..15
            if (m < 4)
                op[16 + m] = g * c1[r];        // cols 16..19 (cols 20..31 padded)
        }
    }
}

extern "C" void kernel_launch(void* const* d_in, const int* in_sizes, int n_in,
                              void* d_out, int out_size, void* d_ws, size_t ws_size,
                              hipStream_t stream) {
    const float* x  = (const float*)d_in[0];
    const float* wg = (const float*)d_in[1];
    const float* We = (const float*)d_in[2];
    const float* be = (const float*)d_in[3];
    float* out = (float*)d_out;

    const int numTokens = in_sizes[0] / HIDDEN;      // 256*4096 = 1,048,576
    const int numTiles  = numTokens >> 4;            // 65,536 wave-tiles
    int blocks = (numTiles + 31) / 32;               // ~4 tiles per wave (8 waves/block)
    if (blocks > 2048) blocks = 2048;
    if (blocks < 1)    blocks = 1;

    moe_top1_wmma_kernel<<<blocks, 256, 0, stream>>>(x, wg, We, be, out, numTokens);
}